// EMAVectorQuantizer_57449482551922
// MI455X (gfx1250) — compile-verified
//
#include <hip/hip_runtime.h>
#include <hip/hip_bf16.h>

// ---------------------------------------------------------------------------
// VQ-VAE vector quantizer for MI455X (gfx1250, wave32, WMMA, LDS-staged,
// software-pipelined fragment loads).
//
// Shapes: z_e (32,256,32,32) f32 ; embed (8192,256) f32
//   N = 32768 flattened vectors, D = 256, K = 8192 codes.
// Outputs (flat, in return order, all f32):
//   z_q_out : 8,388,608   (B,D,H,W)  -- STE forward value == gathered z_q
//   loss    : 1
//   perplex : 1
//   indices : 32,768      (stored as float)
//   dist    : 268,435,456 (N,K)
// ---------------------------------------------------------------------------

typedef __attribute__((ext_vector_type(2))) float v2f;
typedef __attribute__((ext_vector_type(8))) float v8f;

#define VQ_B 32
#define VQ_D 256
#define VQ_HW 1024            // 32*32
#define VQ_N 32768            // B*HW
#define VQ_K 8192
#define VQ_BHW_STRIDE 262144  // D*HW (elements per batch image)

// GEMM tiling
#define T_ROWS 128  // rows per block (contiguous hw, one batch image)
#define T_COLS 64   // cols per block
#define T_DC 64     // D-chunk staged in LDS per iteration
#define T_CHUNKS (VQ_D / T_DC)
#define T_STEPS (T_DC / 4)  // 16 WMMA K-steps per chunk

// d_out element offsets
#define OUT_ZQ 0
#define OUT_LOSS 8388608
#define OUT_PERP 8388609
#define OUT_IDX 8388610
#define OUT_DIST 8421378

// d_ws byte offsets
#define WS_AMIN 0        // u64[N]        262144 B
#define WS_XNORM 262144  // f32[N]        131072 B
#define WS_ENORM 393216  // f32[K]         32768 B
#define WS_CNT 425984    // u32[K]         32768 B
#define WS_IDXI 458752   // i32[N]        131072 B
#define WS_LOSS 589824   // f32[1]

__device__ __forceinline__ unsigned int fkey(float f) {
  unsigned int u = __float_as_uint(f);
  return (u & 0x80000000u) ? ~u : (u | 0x80000000u);  // monotone order-preserving map
}

// --- init scratch ----------------------------------------------------------
__global__ void vq_init(unsigned long long* __restrict__ amin,
                        unsigned int* __restrict__ counts,
                        float* __restrict__ lossAcc) {
  int i = blockIdx.x * blockDim.x + threadIdx.x;
  if (i < VQ_N) amin[i] = ~0ull;
  if (i < VQ_K) counts[i] = 0u;
  if (i == 0) lossAcc[0] = 0.0f;
}

// --- ||x||^2 per flattened z_e row (wave per row) --------------------------
__global__ __launch_bounds__(256) void vq_xnorm(const float* __restrict__ zE,
                                                float* __restrict__ xnorm) {
  const int lane = threadIdx.x & 31;
  const int wave = threadIdx.x >> 5;
  const int row = blockIdx.x * 8 + wave;  // grid = N/8
  const float* p = zE + (size_t)(row >> 10) * VQ_BHW_STRIDE + (row & (VQ_HW - 1));
  float s = 0.0f;
#pragma unroll
  for (int j = 0; j < 8; ++j) {
    float v = p[(size_t)(lane + 32 * j) * VQ_HW];
    s += v * v;
  }
#pragma unroll
  for (int off = 16; off; off >>= 1) s += __shfl_xor(s, off, 32);
  if (lane == 0) xnorm[row] = s;
}

// --- ||e||^2 per codebook row (wave per row, contiguous) -------------------
__global__ __launch_bounds__(256) void vq_enorm(const float* __restrict__ emb,
                                                float* __restrict__ enorm) {
  const int lane = threadIdx.x & 31;
  const int wave = threadIdx.x >> 5;
  const int row = blockIdx.x * 8 + wave;  // grid = K/8
  const float* p = emb + (size_t)row * VQ_D;
  float s = 0.0f;
#pragma unroll
  for (int j = 0; j < 8; ++j) {
    float v = p[lane + 32 * j];
    s += v * v;
  }
#pragma unroll
  for (int off = 16; off; off >>= 1) s += __shfl_xor(s, off, 32);
  if (lane == 0) enorm[row] = s;
}

// --- fused distance GEMM + store + argmin (LDS-staged, pipelined) ----------
// Block = 256 threads (8 waves). Block tile: 128 rows x 64 cols.
// Wave w computes rows [row0 + w*16, +16) x all 64 cols (4 accumulators).
// D is consumed in 4 chunks of 64, staged through 48 KB of LDS with
// coalesced b128 global reads; WMMA fragments come from ds_load_b64,
// issued two K-steps ahead so LDS latency hides under the WMMAs.
// Grid = (K/64, N/128) = (128, 256).
__global__ __launch_bounds__(256) void vq_gemm_dist(
    const float* __restrict__ zE, const float* __restrict__ emb,
    const float* __restrict__ xnorm, const float* __restrict__ enorm,
    float* __restrict__ dist, unsigned long long* __restrict__ amin) {
  // A pairs: As[p][r][q]  (p = d_off/2 in 0..31, r in 0..127, q = d_off&1)
  // B pairs: Bs[p][c][q]  (c in 0..63)
  __shared__ float As[(T_DC / 2) * T_ROWS * 2];  // 8192 f32 = 32 KB
  __shared__ float Bs[(T_DC / 2) * T_COLS * 2];  // 4096 f32 = 16 KB

  const int tid = threadIdx.x;
  const int lane = tid & 31;
  const int wave = tid >> 5;
  const int row0 = blockIdx.y * T_ROWS;
  const int col0 = blockIdx.x * T_COLS;

  const int m = lane & 15;           // M (and N) index for this lane
  const int hi01 = lane >> 4;        // 0: K+{0,1}, 1: K+{2,3}
  const int rLocal = wave * 16 + m;  // this lane's local A row

  // 128 contiguous rows live in one batch image (128 | 1024)
  const int bImg = row0 >> 10;
  const int hw0 = row0 & (VQ_HW - 1);
  const float* aGlob = zE + (size_t)bImg * VQ_BHW_STRIDE + hw0;

  v8f acc[4] = {v8f{}, v8f{}, v8f{}, v8f{}};

  for (int chunk = 0; chunk < T_CHUNKS; ++chunk) {
    const int dBase = chunk * T_DC;
    __syncthreads();

    // --- stage A: 64 d-lines x 128 contiguous floats, coalesced b128 ------
#pragma unroll
    for (int i = 0; i < 8; ++i) {
      const int e4 = tid + i * 256;  // 0..2047 float4 chunks
      const int r4 = (e4 & 31) * 4;  // row offset 0..124
      const int dOff = e4 >> 5;      // 0..63
      const float4 v = *(const float4*)(aGlob + (size_t)(dBase + dOff) * VQ_HW + r4);
      const int p = dOff >> 1, q = dOff & 1;
      As[p * (T_ROWS * 2) + (r4 + 0) * 2 + q] = v.x;
      As[p * (T_ROWS * 2) + (r4 + 1) * 2 + q] = v.y;
      As[p * (T_ROWS * 2) + (r4 + 2) * 2 + q] = v.z;
      As[p * (T_ROWS * 2) + (r4 + 3) * 2 + q] = v.w;
    }
    // --- stage B: 64 codebook rows x 64 contiguous d, coalesced b128 ------
#pragma unroll
    for (int i = 0; i < 4; ++i) {
      const int e4 = tid + i * 256;   // 0..1023 float4 chunks
      const int c = e4 >> 4;          // 0..63
      const int d4o = (e4 & 15) * 4;  // 0..60
      const float4 v =
          *(const float4*)(emb + (size_t)(col0 + c) * VQ_D + dBase + d4o);
      const int p0 = d4o >> 1;
      *(v2f*)&Bs[(p0 + 0) * (T_COLS * 2) + c * 2] = v2f{v.x, v.y};
      *(v2f*)&Bs[(p0 + 1) * (T_COLS * 2) + c * 2] = v2f{v.z, v.w};
    }
    __syncthreads();

    // prefetch next chunk's lines while we compute this one
    if (chunk + 1 < T_CHUNKS) {
      const int dNext = dBase + T_DC;
      __builtin_prefetch(aGlob + (size_t)(dNext + (tid >> 2)) * VQ_HW + (tid & 3) * 32, 0, 3);
      __builtin_prefetch(emb + (size_t)(col0 + (tid >> 2)) * VQ_D + dNext + (tid & 3) * 16, 0, 3);
    }

    // --- hot loop: fragments from LDS, loaded 2 K-steps ahead -------------
    // frag slot for step tl lives in aP[tl&1] / bP[tl&1]
    v2f aP[2];
    v2f bP[2][4];
#pragma unroll
    for (int s = 0; s < 2; ++s) {
      const int p = 2 * s + hi01;
      aP[s] = *(const v2f*)&As[p * (T_ROWS * 2) + rLocal * 2];
#pragma unroll
      for (int cc = 0; cc < 4; ++cc)
        bP[s][cc] = *(const v2f*)&Bs[p * (T_COLS * 2) + (cc * 16 + m) * 2];
    }
#pragma unroll
    for (int tl = 0; tl < T_STEPS; ++tl) {
      const int s = tl & 1;
      // issue loads for step tl+2 before consuming step tl
      v2f aN;
      v2f bN[4];
      if (tl + 2 < T_STEPS) {
        const int pn = 2 * (tl + 2) + hi01;
        aN = *(const v2f*)&As[pn * (T_ROWS * 2) + rLocal * 2];
#pragma unroll
        for (int cc = 0; cc < 4; ++cc)
          bN[cc] = *(const v2f*)&Bs[pn * (T_COLS * 2) + (cc * 16 + m) * 2];
      }
#pragma unroll
      for (int cc = 0; cc < 4; ++cc) {
        acc[cc] = __builtin_amdgcn_wmma_f32_16x16x4_f32(
            /*neg_a=*/false, aP[s], /*neg_b=*/false, bP[s][cc],
            /*c_mod=*/(short)0, acc[cc], /*reuse_a=*/false, /*reuse_b=*/false);
      }
      if (tl + 2 < T_STEPS) {
        aP[s] = aN;
#pragma unroll
        for (int cc = 0; cc < 4; ++cc) bP[s][cc] = bN[cc];
      }
    }
  }

  // Epilogue: C layout -> lane<16: VGPR r = C[M=r][N=lane];
  //                       lane>=16: C[M=r+8][N=lane-16].
  const int waveRow0 = row0 + wave * 16;
  const int hi8 = hi01 * 8;
#pragma unroll
  for (int r = 0; r < 8; ++r) {
    const int row = waveRow0 + r + hi8;
    const float xn = xnorm[row];
    unsigned int bestKey = 0xFFFFFFFFu;
    int bestCol = 0x7FFFFFFF;
#pragma unroll
    for (int cc = 0; cc < 4; ++cc) {
      const int col = col0 + cc * 16 + m;
      const float d = xn + enorm[col] - 2.0f * acc[cc][r];
      dist[(size_t)row * VQ_K + col] = d;
      const unsigned int u = fkey(d);
      if (u < bestKey || (u == bestKey && col < bestCol)) {
        bestKey = u;
        bestCol = col;
      }
    }
    // min-reduce across the 16 lanes of each half-wave (xor masks stay in-half)
#pragma unroll
    for (int off = 1; off < 16; off <<= 1) {
      const unsigned int ok = (unsigned int)__shfl_xor((int)bestKey, off, 32);
      const int oc = __shfl_xor(bestCol, off, 32);
      if (ok < bestKey || (ok == bestKey && oc < bestCol)) {
        bestKey = ok;
        bestCol = oc;
      }
    }
    if (m == 0) {
      const unsigned long long pk =
          ((unsigned long long)bestKey << 32) | (unsigned int)bestCol;
      atomicMin(&amin[row], pk);
    }
  }
}

// --- decode packed argmin -> indices (+ histogram) -------------------------
__global__ void vq_indices(const unsigned long long* __restrict__ amin,
                           int* __restrict__ idxInt, float* __restrict__ idxOut,
                           unsigned int* __restrict__ counts) {
  const int n = blockIdx.x * blockDim.x + threadIdx.x;  // grid covers N
  const unsigned long long pk = amin[n];
  const int k = (int)(pk & 0xFFFFFFFFu);
  idxInt[n] = k;
  idxOut[n] = (float)k;
  atomicAdd(&counts[k], 1u);
}

// --- gather z_q, write (B,D,H,W) output, accumulate commitment loss --------
__global__ __launch_bounds__(256) void vq_gather(
    const float* __restrict__ zE, const float* __restrict__ emb,
    const int* __restrict__ idxInt, float* __restrict__ zqOut,
    float* __restrict__ lossAcc) {
  const size_t tid = (size_t)blockIdx.x * 256 + threadIdx.x;  // 8,388,608 threads
  const int b = (int)(tid >> 18);
  const int d = (int)((tid >> 10) & 255);
  const int hw = (int)(tid & 1023);
  const int n = (b << 10) | hw;
  const int k = idxInt[n];
  const float zq = emb[(size_t)k * VQ_D + d];
  const size_t addr = (size_t)b * VQ_BHW_STRIDE + (size_t)d * VQ_HW + hw;
  const float ze = zE[addr];
  zqOut[addr] = zq;  // straight-through forward value == z_q
  const float diff = zq - ze;
  float s = diff * diff;
#pragma unroll
  for (int off = 16; off; off >>= 1) s += __shfl_xor(s, off, 32);
  if ((threadIdx.x & 31) == 0) atomicAdd(lossAcc, s);
}

// --- finalize: loss scalar + perplexity from usage histogram ---------------
__global__ __launch_bounds__(256) void vq_finalize(
    const unsigned int* __restrict__ counts, const float* __restrict__ lossAcc,
    float* __restrict__ outLoss, float* __restrict__ outPerp) {
  __shared__ float sh[256];
  float s = 0.0f;
  for (int k = threadIdx.x; k < VQ_K; k += 256) {
    const float p = (float)counts[k] / (float)VQ_N;
    s += p * __logf(p + 1e-10f);
  }
  sh[threadIdx.x] = s;
  __syncthreads();
  for (int st = 128; st; st >>= 1) {
    if ((int)threadIdx.x < st) sh[threadIdx.x] += sh[threadIdx.x + st];
    __syncthreads();
  }
  if (threadIdx.x == 0) {
    outLoss[0] = lossAcc[0] * 0.25f / (float)((size_t)VQ_N * VQ_D);
    outPerp[0] = __expf(-sh[0]);
  }
}

extern "C" void kernel_launch(void* const* d_in, const int* in_sizes, int n_in,
                              void* d_out, int out_size, void* d_ws,
                              size_t ws_size, hipStream_t stream) {
  const float* zE = (const float*)d_in[0];   // (32,256,32,32)
  const float* emb = (const float*)d_in[1];  // (8192,256)
  float* out = (float*)d_out;

  char* ws = (char*)d_ws;
  unsigned long long* amin = (unsigned long long*)(ws + WS_AMIN);
  float* xnorm = (float*)(ws + WS_XNORM);
  float* enorm = (float*)(ws + WS_ENORM);
  unsigned int* counts = (unsigned int*)(ws + WS_CNT);
  int* idxInt = (int*)(ws + WS_IDXI);
  float* lossAcc = (float*)(ws + WS_LOSS);

  float* zqOut = out + OUT_ZQ;
  float* outLoss = out + OUT_LOSS;
  float* outPerp = out + OUT_PERP;
  float* idxOut = out + OUT_IDX;
  float* dist = out + OUT_DIST;

  vq_init<<<VQ_N / 256, 256, 0, stream>>>(amin, counts, lossAcc);
  vq_xnorm<<<VQ_N / 8, 256, 0, stream>>>(zE, xnorm);
  vq_enorm<<<VQ_K / 8, 256, 0, stream>>>(emb, enorm);
  vq_gemm_dist<<<dim3(VQ_K / T_COLS, VQ_N / T_ROWS), 256, 0, stream>>>(
      zE, emb, xnorm, enorm, dist, amin);
  vq_indices<<<VQ_N / 256, 256, 0, stream>>>(amin, idxInt, idxOut, counts);
  vq_gather<<<(VQ_N * VQ_D) / 256, 256, 0, stream>>>(zE, emb, idxInt, zqOut,
                                                     lossAcc);
  vq_finalize<<<1, 256, 0, stream>>>(counts, lossAcc, outLoss, outPerp);
}